// MeanIoU_36395552866650
// MI455X (gfx1250) — compile-verified
//
#include <hip/hip_runtime.h>
#include <stdint.h>

#define NC     21
#define TPB    256
#define NBLK   1024
#define WAVES  (TPB / 32)

// 4 x i32 vector type matching the builtin's expected pointee for b128
typedef int v4i __attribute__((ext_vector_type(4)));
typedef __attribute__((address_space(1))) v4i* as1_v4i_ptr;
typedef __attribute__((address_space(3))) v4i* as3_v4i_ptr;

// ---------------- CDNA5 async global->LDS helpers ----------------
// Builtin confirmed present by round-1 diagnostic; signature:
//   (v4i AS1* gaddr, v4i AS3* lds, imm offset, imm cpol)
__device__ __forceinline__ void async_copy_b128(const void* gp, void* lp) {
#if defined(__gfx1250__) && __has_builtin(__builtin_amdgcn_global_load_async_to_lds_b128)
  __builtin_amdgcn_global_load_async_to_lds_b128(
      (as1_v4i_ptr)(uintptr_t)gp,
      (as3_v4i_ptr)(uint32_t)(uintptr_t)lp,   // low 32 bits of generic ptr = LDS offset
      0, 0);
#else
  unsigned l = (unsigned)(uintptr_t)lp;
  unsigned long long g = (unsigned long long)(uintptr_t)gp;
  asm volatile("global_load_async_to_lds_b128 %0, %1, off" :: "v"(l), "v"(g) : "memory");
#endif
}

__device__ __forceinline__ void wait_async_le2() {
  asm volatile("s_wait_asynccnt 2" ::: "memory");
}
__device__ __forceinline__ void wait_async_0() {
  asm volatile("s_wait_asynccnt 0" ::: "memory");
}

// ---------------- kernels ----------------
__global__ void miou_zero(unsigned* __restrict__ g) {
  g[threadIdx.x] = 0u;   // launched with 64 threads; bins 0..62 used
}

__global__ __launch_bounds__(TPB) void miou_hist(const int* __restrict__ yt,
                                                 const int* __restrict__ yp,
                                                 unsigned* __restrict__ gbins,
                                                 int n) {
  // Double-buffered staging tiles (2 bufs x {true,pred} x 256 lanes x 16B = 16 KB)
  __shared__ int4 stage[2][2][TPB];
  // Per-wave sub-histograms: [0..31] packed true/intersection (by yt), [32..63] pred (by yp)
  __shared__ unsigned hist[WAVES][64];

  const unsigned tid = threadIdx.x;
  const unsigned wv  = tid >> 5;                       // wave32 index

  for (unsigned i = tid; i < WAVES * 64u; i += TPB) (&hist[0][0])[i] = 0u;
  __syncthreads();

  const int nvec  = n >> 2;                            // int4 count
  const int vstep = NBLK * TPB;                        // int4 per grid pass
  const int iters = nvec / vstep;                      // = 8 for 32*512*512
  const int vbase = (int)(blockIdx.x * TPB + tid);

  const int4* t4p = (const int4*)yt;
  const int4* p4p = (const int4*)yp;

  if (iters > 0) {
    async_copy_b128(t4p + vbase, &stage[0][0][tid]);
    async_copy_b128(p4p + vbase, &stage[0][1][tid]);
  }

  for (int i = 0; i < iters; ++i) {
    if (i + 1 < iters) {
      const int vidx = vbase + (i + 1) * vstep;
      const int nb   = (i + 1) & 1;
      async_copy_b128(t4p + vidx, &stage[nb][0][tid]);
      async_copy_b128(p4p + vidx, &stage[nb][1][tid]);
      wait_async_le2();                                // iter i's 2 loads complete (in-order)
    } else {
      wait_async_0();
    }
    const int b = i & 1;
    int4 t = stage[b][0][tid];
    int4 p = stage[b][1][tid];

    // packed: low16 = count_true, high16 = intersection (match folded in)
    atomicAdd(&hist[wv][(unsigned)t.x & 31u], 1u + ((t.x == p.x) ? 0x10000u : 0u));
    atomicAdd(&hist[wv][32u + ((unsigned)p.x & 31u)], 1u);
    atomicAdd(&hist[wv][(unsigned)t.y & 31u], 1u + ((t.y == p.y) ? 0x10000u : 0u));
    atomicAdd(&hist[wv][32u + ((unsigned)p.y & 31u)], 1u);
    atomicAdd(&hist[wv][(unsigned)t.z & 31u], 1u + ((t.z == p.z) ? 0x10000u : 0u));
    atomicAdd(&hist[wv][32u + ((unsigned)p.z & 31u)], 1u);
    atomicAdd(&hist[wv][(unsigned)t.w & 31u], 1u + ((t.w == p.w) ? 0x10000u : 0u));
    atomicAdd(&hist[wv][32u + ((unsigned)p.w & 31u)], 1u);
  }

  // Tail (n not divisible by grid chunk) — never taken for 32*512*512, kept for safety
  for (long long idx = (long long)iters * vstep * 4 + (long long)blockIdx.x * TPB + tid;
       idx < (long long)n; idx += (long long)NBLK * TPB) {
    int tv = yt[idx], pv = yp[idx];
    atomicAdd(&hist[wv][(unsigned)tv & 31u], 1u + ((tv == pv) ? 0x10000u : 0u));
    atomicAdd(&hist[wv][32u + ((unsigned)pv & 31u)], 1u);
  }

  __syncthreads();

  // Cross-wave reduce + merge to global bins:
  // gbins[0..20]=count_true, [21..41]=intersection, [42..62]=count_pred
  if (tid < 64u) {
    unsigned s = 0u;
#pragma unroll
    for (int c = 0; c < WAVES; ++c) s += hist[c][tid];
    if (tid < NC) {
      atomicAdd(&gbins[tid],      s & 0xFFFFu);        // count_true
      atomicAdd(&gbins[NC + tid], s >> 16);            // intersection
    } else if (tid >= 32u && tid < 32u + NC) {
      atomicAdd(&gbins[2 * NC + (tid - 32u)], s);      // count_pred
    }
  }
}

__global__ void miou_final(const unsigned* __restrict__ g, float* __restrict__ out) {
  __shared__ float part[NC];
  const unsigned c = threadIdx.x;
  if (c < NC) {
    const unsigned ct = g[c];
    const unsigned ci = g[NC + c];
    const unsigned cp = g[2 * NC + c];
    const unsigned un = ct + cp - ci;
    part[c] = (float)ci / (float)un;
  }
  __syncthreads();
  if (c == 0) {
    float s = 0.0f;
    for (int i = 0; i < NC; ++i) s += part[i];
    out[0] = s / (float)NC;
  }
}

// ---------------- host entry ----------------
extern "C" void kernel_launch(void* const* d_in, const int* in_sizes, int n_in,
                              void* d_out, int out_size, void* d_ws, size_t ws_size,
                              hipStream_t stream) {
  (void)n_in; (void)out_size; (void)ws_size;
  const int* yt = (const int*)d_in[0];
  const int* yp = (const int*)d_in[1];
  unsigned* gbins = (unsigned*)d_ws;                   // 63 u32 bins (252 B)
  float* out = (float*)d_out;
  const int n = in_sizes[0];

  miou_zero <<<1, 64, 0, stream>>>(gbins);
  miou_hist <<<NBLK, TPB, 0, stream>>>(yt, yp, gbins, n);
  miou_final<<<1, 32, 0, stream>>>(gbins, out);
}